// QuantumKernelRegressor_65481071407984
// MI455X (gfx1250) — compile-verified
//
#include <hip/hip_runtime.h>
#include <hip/hip_bf16.h>

typedef __attribute__((ext_vector_type(2))) float v2f;
typedef __attribute__((ext_vector_type(8))) float v8f;

// ---------------------------------------------------------------------------
// Kernel 1: featurize rows: x[r][0..3] -> phi[r][0..15]
// phi_S = prod_{d in S} sin(x_d/2) * prod_{d notin S} cos(x_d/2)
// ---------------------------------------------------------------------------
__global__ void qkr_features(const float* __restrict__ x,
                             float* __restrict__ phi, int rows) {
  int r = blockIdx.x * blockDim.x + threadIdx.x;
  if (r >= rows) return;
  float c[4], s[4];
#pragma unroll
  for (int d = 0; d < 4; ++d) {
    float h = 0.5f * x[r * 4 + d];
    sincosf(h, &s[d], &c[d]);
  }
#pragma unroll
  for (int S = 0; S < 16; ++S) {
    float f = ((S & 1) ? s[0] : c[0]) * ((S & 2) ? s[1] : c[1]) *
              ((S & 4) ? s[2] : c[2]) * ((S & 8) ? s[3] : c[3]);
    phi[r * 16 + S] = f;
  }
}

// ---------------------------------------------------------------------------
// Kernel 2: one wave per (row-tile, N-chunk).
//   C(16x16) = PhiX_tile(16x16) @ PhiT_tile^T(16x16) via 4x V_WMMA_F32_16X16X4_F32
//   acc[m] += |C[m][n]| * alpha[n]  (accumulated over the chunk's N-tiles)
// Partials written to ws (no atomics -> deterministic).
// ---------------------------------------------------------------------------
__global__ __launch_bounds__(256) void qkr_tile(
    const float* __restrict__ phiX, const float* __restrict__ phiT,
    const float* __restrict__ alpha, float* __restrict__ partial,
    int M, int nchunk, int tilesPerChunk) {
  const int wavesPerBlock = 256 / 32;
  int w = blockIdx.x * wavesPerBlock + (threadIdx.x >> 5);
  int rowTile = w / nchunk;
  int chunk   = w % nchunk;
  int m0 = rowTile * 16;

  int lane = threadIdx.x & 31;
  int half = lane >> 4;    // 0: K 0,1 of each 4 / C rows 0..7, 1: K 2,3 / C rows 8..15 (per ISA layout)
  int l16  = lane & 15;

  // A fragment: PhiX rows m0..m0+15, all 16 features (4 K-chunks of 4).
  // 32-bit A 16x4 layout: lane<16 holds K = k0,k0+1 ; lane>=16 holds K = k0+2,k0+3
  v2f a[4];
#pragma unroll
  for (int k = 0; k < 4; ++k) {
    const float* p = &phiX[(m0 + l16) * 16 + k * 4 + 2 * half];
    a[k] = *(const v2f*)p;
  }

  float acc[8];
#pragma unroll
  for (int v = 0; v < 8; ++v) acc[v] = 0.0f;

  int t0 = chunk * tilesPerChunk;
  for (int t = 0; t < tilesPerChunk; ++t) {
    int n0 = (t0 + t) * 16;

    // B fragment: PhiT rows n0..n0+15 (B is K x N; lane indexes N, VGPRs index K)
    v2f b[4];
#pragma unroll
    for (int k = 0; k < 4; ++k) {
      const float* p = &phiT[(n0 + l16) * 16 + k * 4 + 2 * half];
      b[k] = *(const v2f*)p;
    }
    float alpha_n = alpha[n0 + l16];

    v8f c = {};
#pragma unroll
    for (int k = 0; k < 4; ++k) {
      // 8 args: (neg_a, A, neg_b, B, c_mod, C, reuse_a, reuse_b)
      c = __builtin_amdgcn_wmma_f32_16x16x4_f32(
          false, a[k], false, b[k], (short)0, c, false, false);
    }

    // C layout: VGPR v, lane L -> row m0 + v + 8*(L>=16), col n0 + (L%16)
#pragma unroll
    for (int v = 0; v < 8; ++v) acc[v] += __builtin_fabsf(c[v]) * alpha_n;
  }

  // Reduce each acc[v] across the 16 lanes of this half (sum over n in tile col set)
#pragma unroll
  for (int v = 0; v < 8; ++v) {
#pragma unroll
    for (int m = 1; m < 16; m <<= 1) acc[v] += __shfl_xor(acc[v], m, 16);
  }

  if (l16 == 0) {
#pragma unroll
    for (int v = 0; v < 8; ++v)
      partial[(size_t)chunk * M + m0 + half * 8 + v] = acc[v];
  }
}

// ---------------------------------------------------------------------------
// Kernel 3: out[m] = sum_chunks partial[chunk][m]
// ---------------------------------------------------------------------------
__global__ void qkr_reduce(const float* __restrict__ partial,
                           float* __restrict__ out, int M, int nchunk) {
  int m = blockIdx.x * blockDim.x + threadIdx.x;
  if (m >= M) return;
  float s = 0.0f;
  for (int c = 0; c < nchunk; ++c) s += partial[(size_t)c * M + m];
  out[m] = s;
}

extern "C" void kernel_launch(void* const* d_in, const int* in_sizes, int n_in,
                              void* d_out, int out_size, void* d_ws, size_t ws_size,
                              hipStream_t stream) {
  const float* X     = (const float*)d_in[0];  // [M,4]
  const float* T     = (const float*)d_in[1];  // [N,4]
  const float* alpha = (const float*)d_in[2];  // [N]
  float* out = (float*)d_out;

  int M = in_sizes[0] / 4;
  int N = in_sizes[1] / 4;

  float* phiX    = (float*)d_ws;              // M*16 floats
  float* phiT    = phiX + (size_t)M * 16;     // N*16 floats
  float* partial = phiT + (size_t)N * 16;     // nchunk*M floats

  int nTiles = N / 16;
  int nchunk = 64;
  while (nchunk > 1 && (nTiles % nchunk)) nchunk >>= 1;
  int tilesPerChunk = nTiles / nchunk;

  qkr_features<<<(M + 255) / 256, 256, 0, stream>>>(X, phiX, M);
  qkr_features<<<(N + 255) / 256, 256, 0, stream>>>(T, phiT, N);

  int waves  = (M / 16) * nchunk;             // 128 * 64 = 8192 waves
  int blocks = waves / (256 / 32);
  qkr_tile<<<blocks, 256, 0, stream>>>(phiX, phiT, alpha, partial,
                                       M, nchunk, tilesPerChunk);

  qkr_reduce<<<(M + 255) / 256, 256, 0, stream>>>(partial, out, M, nchunk);
}